// PointTransformerCls_22909355556979
// MI455X (gfx1250) — compile-verified
//
#include <hip/hip_runtime.h>
#include <hip/hip_bf16.h>
#include <math.h>

// ============================================================================
// PointTransformer classifier forward for MI455X (gfx1250, wave32).
// Dense linear layers run through fused WMMA f32 16x16x4 GEMMs
// (bias + BatchNorm + ReLU epilogue). Hot path: branch-free aligned kernel,
// one wave = 16x(16*NT) output strip, A-frag reused across NT WMMAs.
// Irregular stages (KNN top-16, FPS, gathers, softmax, pooling) are VALU.
// ============================================================================

typedef float v2f __attribute__((ext_vector_type(2)));
typedef float v8f __attribute__((ext_vector_type(8)));
typedef v2f v2f_u __attribute__((aligned(4)));   // DWORD-aligned pair load

#define KNBR 16
#define BN_EPS 1e-5f

// ---------------------------------------------------------------------------
// Aligned fused GEMM: Y[M,N] = X[M,K] @ W[K,N] (+bias)(+BN)(+ReLU)
// Requires M%16==0 and N%(16*NT)==0; K arbitrary (tail via clamp+select).
// A-frag (16x4 f32): lanes 0-15 -> K=k0,k0+1 ; lanes 16-31 -> k0+2,k0+3
// B-frag (4x16 f32): lane half selects K pair, lane&15 = column
// C/D (16x16 f32, 8 VGPR): vgpr r -> rows r (lanes 0-15) / r+8 (lanes 16-31)
// ---------------------------------------------------------------------------
template <int NT>
__global__ void __launch_bounds__(256) gemm_wmma_aligned(
    const float* __restrict__ X, const float* __restrict__ W,
    const float* __restrict__ bias,
    const float* __restrict__ bng, const float* __restrict__ bnb,
    const float* __restrict__ bnm, const float* __restrict__ bnv,
    int relu, float* __restrict__ Y, int M, int N, int K)
{
  int lane  = threadIdx.x & 31;
  int lan15 = lane & 15;
  int khalf = lane >> 4;
  int Mt = M >> 4;
  int Ng = N / (16 * NT);
  long tile = (long)blockIdx.x * (blockDim.x >> 5) + (threadIdx.x >> 5);
  if (tile >= (long)Mt * Ng) return;            // whole-wave uniform exit
  int tm = (int)(tile % Mt), tg = (int)(tile / Mt);
  int mrow = tm * 16 + lan15;
  int col0 = tg * (16 * NT) + lan15;

  const float* ap = X + (size_t)mrow * K + khalf * 2;
  const float* wp = W + col0 + (size_t)(khalf * 2) * N;

  v8f acc[NT];
#pragma unroll
  for (int t = 0; t < NT; ++t) acc[t] = {};

  int Kmain = K & ~3;
  for (int k0 = 0; k0 < Kmain; k0 += 4) {
    v2f a = *(const v2f_u*)ap;                  // one b64 load
    ap += 4;
#pragma unroll
    for (int t = 0; t < NT; ++t) {
      v2f b;
      b.x = wp[t * 16];
      b.y = wp[t * 16 + N];
      acc[t] = __builtin_amdgcn_wmma_f32_16x16x4_f32(
          false, a, false, b, (short)0, acc[t], false, false);
    }
    wp += (size_t)4 * N;
  }
  if (Kmain < K) {                              // K tail, branch-free
    int ks  = Kmain + khalf * 2;
    int k0c = (ks     < K) ? ks     : K - 1;
    int k1c = (ks + 1 < K) ? ks + 1 : K - 1;
    const float* xr = X + (size_t)mrow * K;
    float a0 = xr[k0c], a1 = xr[k1c];
    v2f a;
    a.x = (ks     < K) ? a0 : 0.f;
    a.y = (ks + 1 < K) ? a1 : 0.f;
#pragma unroll
    for (int t = 0; t < NT; ++t) {
      const float* wc = W + col0 + t * 16;
      float b0 = wc[(size_t)k0c * N], b1 = wc[(size_t)k1c * N];
      v2f b;
      b.x = (ks     < K) ? b0 : 0.f;
      b.y = (ks + 1 < K) ? b1 : 0.f;
      acc[t] = __builtin_amdgcn_wmma_f32_16x16x4_f32(
          false, a, false, b, (short)0, acc[t], false, false);
    }
  }
  // epilogue (unconditional: M,N aligned)
#pragma unroll
  for (int t = 0; t < NT; ++t) {
    int ncol = col0 + t * 16;
    float bv = bias ? bias[ncol] : 0.f;
    float g = 1.f, b2 = 0.f, mu = 0.f, inv = 1.f;
    if (bng) { g = bng[ncol]; b2 = bnb[ncol]; mu = bnm[ncol];
               inv = rsqrtf(bnv[ncol] + BN_EPS); }
#pragma unroll
    for (int r = 0; r < 8; ++r) {
      int mr = tm * 16 + khalf * 8 + r;
      float val = acc[t][r] + bv;
      if (bng) val = (val - mu) * inv * g + b2;
      if (relu) val = fmaxf(val, 0.f);
      Y[(size_t)mr * N + ncol] = val;
    }
  }
}

// ---------------------------------------------------------------------------
// Generic fused GEMM (arbitrary M,N,K) -- only used for tiny head GEMMs.
// Branch-free: clamped indices + selects, no guarded loads.
// ---------------------------------------------------------------------------
__global__ void __launch_bounds__(256) gemm_wmma_generic(
    const float* __restrict__ X, const float* __restrict__ W,
    const float* __restrict__ bias,
    const float* __restrict__ bng, const float* __restrict__ bnb,
    const float* __restrict__ bnm, const float* __restrict__ bnv,
    int relu, float* __restrict__ Y, int M, int N, int K)
{
  int lane  = threadIdx.x & 31;
  int lan15 = lane & 15;
  int khalf = lane >> 4;
  int Mt = (M + 15) >> 4, Nt = (N + 15) >> 4;
  long tile = (long)blockIdx.x * (blockDim.x >> 5) + (threadIdx.x >> 5);
  if (tile >= (long)Mt * Nt) return;
  int tm = (int)(tile % Mt), tn = (int)(tile / Mt);
  int mrow = tm * 16 + lan15;
  int ncol = tn * 16 + lan15;
  int mrc = (mrow < M) ? mrow : M - 1;
  int ncc = (ncol < N) ? ncol : N - 1;
  bool mok = mrow < M, nok = ncol < N;
  const float* xr = X + (size_t)mrc * K;

  v8f acc = {};
  int Kmain = K & ~3;
  for (int k0 = 0; k0 < Kmain; k0 += 4) {
    int ks = k0 + khalf * 2;
    float a0 = xr[ks], a1 = xr[ks + 1];
    float b0 = W[(size_t)ks * N + ncc], b1 = W[(size_t)(ks + 1) * N + ncc];
    v2f a, b;
    a.x = mok ? a0 : 0.f;  a.y = mok ? a1 : 0.f;
    b.x = nok ? b0 : 0.f;  b.y = nok ? b1 : 0.f;
    acc = __builtin_amdgcn_wmma_f32_16x16x4_f32(
        false, a, false, b, (short)0, acc, false, false);
  }
  if (Kmain < K) {
    int ks  = Kmain + khalf * 2;
    int k0c = (ks     < K) ? ks     : K - 1;
    int k1c = (ks + 1 < K) ? ks + 1 : K - 1;
    float a0 = xr[k0c], a1 = xr[k1c];
    float b0 = W[(size_t)k0c * N + ncc], b1 = W[(size_t)k1c * N + ncc];
    v2f a, b;
    a.x = (mok && ks     < K) ? a0 : 0.f;
    a.y = (mok && ks + 1 < K) ? a1 : 0.f;
    b.x = (nok && ks     < K) ? b0 : 0.f;
    b.y = (nok && ks + 1 < K) ? b1 : 0.f;
    acc = __builtin_amdgcn_wmma_f32_16x16x4_f32(
        false, a, false, b, (short)0, acc, false, false);
  }
  if (nok) {
    float bv = bias ? bias[ncol] : 0.f;
    float g = 1.f, b2 = 0.f, mu = 0.f, inv = 1.f;
    if (bng) { g = bng[ncol]; b2 = bnb[ncol]; mu = bnm[ncol];
               inv = rsqrtf(bnv[ncol] + BN_EPS); }
#pragma unroll
    for (int r = 0; r < 8; ++r) {
      int mr = tm * 16 + khalf * 8 + r;
      if (mr < M) {
        float val = acc[r] + bv;
        if (bng) val = (val - mu) * inv * g + b2;
        if (relu) val = fmaxf(val, 0.f);
        Y[(size_t)mr * N + ncol] = val;
      }
    }
  }
}

// ---------------------------------------------------------------------------
// mask[b,n] = any(x[b,n,:] != 0)
// ---------------------------------------------------------------------------
__global__ void mask_kernel(const float* __restrict__ x, unsigned char* __restrict__ mask, long BN)
{
  long t = (long)blockIdx.x * blockDim.x + threadIdx.x;
  if (t >= BN) return;
  const float* p = x + t * 3;
  mask[t] = (p[0] != 0.f || p[1] != 0.f || p[2] != 0.f) ? 1 : 0;
}

// ---------------------------------------------------------------------------
// KNN: 16 nearest valid source points per query (squared L2, low-idx ties).
// ---------------------------------------------------------------------------
__global__ void knn_kernel(const float* __restrict__ xyz, const float* __restrict__ qry,
                           const unsigned char* __restrict__ mask,
                           int* __restrict__ outIdx, int B, int Nsrc, int M)
{
  long t = (long)blockIdx.x * blockDim.x + threadIdx.x;
  if (t >= (long)B * M) return;
  int b = (int)(t / M);
  const float* xb = xyz + (size_t)b * Nsrc * 3;
  float qx = qry[t * 3 + 0], qy = qry[t * 3 + 1], qz = qry[t * 3 + 2];
  float bd[KNBR]; int bi[KNBR];
  int kk = (Nsrc < KNBR) ? Nsrc : KNBR;
#pragma unroll
  for (int i = 0; i < KNBR; ++i) { bd[i] = __builtin_inff(); bi[i] = 0; }
  for (int j = 0; j < Nsrc; ++j) {
    float px = xb[3 * j], py = xb[3 * j + 1], pz = xb[3 * j + 2];
    bool valid = (px != 0.f || py != 0.f || pz != 0.f);
    if (mask) valid = valid && (mask[(size_t)b * Nsrc + j] != 0);
    if (!valid) continue;
    float dx = qx - px, dy = qy - py, dz = qz - pz;
    float d2 = dx * dx + dy * dy + dz * dz;
    if (d2 < bd[kk - 1]) {
      int pos = kk - 1;
      while (pos > 0 && bd[pos - 1] > d2) {
        bd[pos] = bd[pos - 1]; bi[pos] = bi[pos - 1]; --pos;
      }
      bd[pos] = d2; bi[pos] = j;
    }
  }
  for (int i = kk; i < KNBR; ++i) bi[i] = bi[i - kk];
  int* o = outIdx + t * KNBR;
#pragma unroll
  for (int i = 0; i < KNBR; ++i) o[i] = bi[i];
}

// ---------------------------------------------------------------------------
// Farthest point sampling: one block per batch, start at index 0.
// ---------------------------------------------------------------------------
__global__ void fps_kernel(const float* __restrict__ xyz, int* __restrict__ out,
                           int N, int npoint)
{
  __shared__ float sdist[2048];
  __shared__ float rv[256];
  __shared__ int   ri[256];
  __shared__ int   sfar;
  int b = blockIdx.x, tid = threadIdx.x;
  const float* xb = xyz + (size_t)b * N * 3;
  for (int i = tid; i < N; i += 256) sdist[i] = 1e10f;
  if (tid == 0) sfar = 0;
  __syncthreads();
  for (int it = 0; it < npoint; ++it) {
    int far = sfar;
    if (tid == 0) out[(size_t)b * npoint + it] = far;
    float cx = xb[3 * far], cy = xb[3 * far + 1], cz = xb[3 * far + 2];
    float bv = -1.f; int bj = 0;
    for (int j = tid; j < N; j += 256) {
      float dx = xb[3 * j] - cx, dy = xb[3 * j + 1] - cy, dz = xb[3 * j + 2] - cz;
      float d = dx * dx + dy * dy + dz * dz;
      float nd = fminf(sdist[j], d);
      sdist[j] = nd;
      if (nd > bv) { bv = nd; bj = j; }
    }
    rv[tid] = bv; ri[tid] = bj;
    __syncthreads();
    for (int s = 128; s > 0; s >>= 1) {
      if (tid < s) {
        if (rv[tid + s] > rv[tid] ||
            (rv[tid + s] == rv[tid] && ri[tid + s] < ri[tid])) {
          rv[tid] = rv[tid + s]; ri[tid] = ri[tid + s];
        }
      }
      __syncthreads();
    }
    if (tid == 0) sfar = ri[0];
    __syncthreads();
  }
}

// p2[b,m,:] = p1[b, idx[b,m], :]
__global__ void gather_pts_kernel(const float* __restrict__ p1, const int* __restrict__ idx,
                                  float* __restrict__ p2, int N, int M, long total)
{
  long t = (long)blockIdx.x * blockDim.x + threadIdx.x;
  if (t >= total) return;           // total = B*M*3
  int d = (int)(t % 3); long t2 = t / 3;
  int m = (int)(t2 % M); int b = (int)(t2 / M);
  int j = idx[(size_t)b * M + m];
  p2[t] = p1[((size_t)b * N + j) * 3 + d];
}

// rel[b,n,k,:] = p[b,n,:] - p[b, idx[b,n,k], :]
__global__ void rel_kernel(const float* __restrict__ p, const int* __restrict__ idx,
                           float* __restrict__ rel, int N, long total)
{
  long t = (long)blockIdx.x * blockDim.x + threadIdx.x;
  if (t >= total) return;           // total = B*N*K
  long t2 = t / KNBR;
  int n = (int)(t2 % N); int b = (int)(t2 / N);
  int j = idx[t];
  const float* pn = p + ((size_t)b * N + n) * 3;
  const float* pj = p + ((size_t)b * N + j) * 3;
  float* o = rel + t * 3;
  o[0] = pn[0] - pj[0]; o[1] = pn[1] - pj[1]; o[2] = pn[2] - pj[2];
}

// a[b,n,k,c] = relu(BN(q[b,n,c] - kf[b,idx,c] + r[b,n,k,c]))   (at_bn1 fused)
__global__ void abuild_kernel(const float* __restrict__ q, const float* __restrict__ kf,
                              const float* __restrict__ r, const int* __restrict__ idx,
                              const float* __restrict__ g, const float* __restrict__ bb,
                              const float* __restrict__ mu, const float* __restrict__ vv,
                              float* __restrict__ a, int N, int C, long total)
{
  long t = (long)blockIdx.x * blockDim.x + threadIdx.x;
  if (t >= total) return;           // total = B*N*K*C
  int c = (int)(t % C); long t2 = t / C;
  int k = (int)(t2 % KNBR); long t3 = t2 / KNBR;
  int n = (int)(t3 % N); int b = (int)(t3 / N);
  int j = idx[((size_t)b * N + n) * KNBR + k];
  float val = q[((size_t)b * N + n) * C + c] - kf[((size_t)b * N + j) * C + c] + r[t];
  val = (val - mu[c]) * rsqrtf(vv[c] + BN_EPS) * g[c] + bb[c];
  a[t] = fmaxf(val, 0.f);
}

// softmax over K per (b,n,c), then out[b,n,c] = sum_k w_k * (v[b,idx,c] + r[b,n,k,c])
__global__ void softmax_agg_kernel(const float* __restrict__ a, const float* __restrict__ v,
                                   const float* __restrict__ r, const int* __restrict__ idx,
                                   const unsigned char* __restrict__ mask,
                                   float* __restrict__ out, int N, int C, long total)
{
  long t = (long)blockIdx.x * blockDim.x + threadIdx.x;
  if (t >= total) return;           // total = B*N*C
  int c = (int)(t % C); long t2 = t / C;
  int n = (int)(t2 % N); int b = (int)(t2 / N);
  size_t base = ((size_t)b * N + n) * KNBR * (size_t)C + c;
  bool valid = mask ? (mask[(size_t)b * N + n] != 0) : true;
  float av[KNBR];
#pragma unroll
  for (int k = 0; k < KNBR; ++k) av[k] = valid ? a[base + (size_t)k * C] : 0.f;
  float mx = av[0];
#pragma unroll
  for (int k = 1; k < KNBR; ++k) mx = fmaxf(mx, av[k]);
  float s = 0.f;
#pragma unroll
  for (int k = 0; k < KNBR; ++k) { av[k] = expf(av[k] - mx); s += av[k]; }
  float inv = 1.f / s;              // s >= 1, no NaN
  float acc = 0.f;
  const int* ib = idx + ((size_t)b * N + n) * KNBR;
#pragma unroll
  for (int k = 0; k < KNBR; ++k) {
    int j = ib[k];
    acc += av[k] * inv * (v[((size_t)b * N + j) * C + c] + r[base + (size_t)k * C]);
  }
  out[t] = acc;
}

// h = relu(y + x)
__global__ void residual_relu_kernel(const float* __restrict__ y, const float* __restrict__ x,
                                     float* __restrict__ h, long total)
{
  long t = (long)blockIdx.x * blockDim.x + threadIdx.x;
  if (t >= total) return;
  h[t] = fmaxf(y[t] + x[t], 0.f);
}

// f[b,m,k,0:3] = p1[idx]-p2 ; f[b,m,k,3:3+C] = x1[idx]
__global__ void tdfeat_kernel(const float* __restrict__ p1, const float* __restrict__ p2,
                              const float* __restrict__ x1, const int* __restrict__ idx,
                              float* __restrict__ f, int Nsrc, int M, int C, long total)
{
  long t = (long)blockIdx.x * blockDim.x + threadIdx.x;
  if (t >= total) return;           // total = B*M*K*(3+C)
  int F = 3 + C;
  int c = (int)(t % F); long t2 = t / F;
  int k = (int)(t2 % KNBR); long t3 = t2 / KNBR;
  int m = (int)(t3 % M); int b = (int)(t3 / M);
  int j = idx[((size_t)b * M + m) * KNBR + k];
  float val;
  if (c < 3) val = p1[((size_t)b * Nsrc + j) * 3 + c] - p2[((size_t)b * M + m) * 3 + c];
  else       val = x1[((size_t)b * Nsrc + j) * C + (c - 3)];
  f[t] = val;
}

// out[b,m,c] = max_k in[b,m,k,c]
__global__ void maxpool_kernel(const float* __restrict__ in, float* __restrict__ out,
                               int M, int C, long total)
{
  long t = (long)blockIdx.x * blockDim.x + threadIdx.x;
  if (t >= total) return;           // total = B*M*C
  int c = (int)(t % C); long t2 = t / C;
  int m = (int)(t2 % M); int b = (int)(t2 / M);
  size_t base = ((size_t)b * M + m) * KNBR * (size_t)C + c;
  float mx = -__builtin_inff();
#pragma unroll
  for (int k = 0; k < KNBR; ++k) mx = fmaxf(mx, in[base + (size_t)k * C]);
  out[t] = mx;
}

// out[b,c] = sum_n h[b,n,c]*w / max(sum w, 1)
__global__ void avgpool_kernel(const float* __restrict__ h, const unsigned char* __restrict__ mask,
                               float* __restrict__ out, int N, int C, long total)
{
  long t = (long)blockIdx.x * blockDim.x + threadIdx.x;
  if (t >= total) return;           // total = B*C
  int c = (int)(t % C); int b = (int)(t / C);
  float s = 0.f, ws = 0.f;
  for (int n = 0; n < N; ++n) {
    float w = mask ? (mask[(size_t)b * N + n] ? 1.f : 0.f) : 1.f;
    s += h[((size_t)b * N + n) * C + c] * w;
    ws += w;
  }
  out[t] = s / fmaxf(ws, 1.f);
}

// ============================================================================
// Host side
// ============================================================================

struct Lin { const float* w; const float* b; };
struct Bn  { const float* g; const float* b; const float* m; const float* v; };
struct Layer { Lin q, k, v, pe1; Bn pe_bn1; Lin pe2; Bn at_bn1; Lin at1; Bn at_bn2; Lin at2; };
struct Block { Lin l1; Bn bn1; Layer tr; Lin l2; Bn bn; };
struct Td    { Lin l1; Bn bn1; Lin l2; Bn bn2; };

struct Cursor {
  void* const* d; int i;
  const float* nf() { return (const float*)d[i++]; }
};
static Lin   getLin(Cursor& c, bool bias) { Lin l; l.w = c.nf(); l.b = bias ? c.nf() : nullptr; return l; }
static Bn    getBn(Cursor& c)  { Bn b; b.g = c.nf(); b.b = c.nf(); b.m = c.nf(); b.v = c.nf(); return b; }
static Layer getLayer(Cursor& c) {
  Layer L; L.q = getLin(c, true); L.k = getLin(c, true); L.v = getLin(c, true);
  L.pe1 = getLin(c, false); L.pe_bn1 = getBn(c); L.pe2 = getLin(c, true);
  L.at_bn1 = getBn(c); L.at1 = getLin(c, false); L.at_bn2 = getBn(c); L.at2 = getLin(c, true);
  return L;
}
static Block getBlock(Cursor& c) {
  Block b; b.l1 = getLin(c, false); b.bn1 = getBn(c); b.tr = getLayer(c);
  b.l2 = getLin(c, false); b.bn = getBn(c); return b;
}
static Td getTd(Cursor& c) {
  Td t; t.l1 = getLin(c, false); t.bn1 = getBn(c); t.l2 = getLin(c, false); t.bn2 = getBn(c);
  return t;
}

static inline long cdiv(long a, long b) { return (a + b - 1) / b; }

static void gemm(hipStream_t s, const float* X, const float* W, const float* bias,
                 const Bn* bn, int relu, float* Y, long M, int N, int K)
{
  const float* g  = bn ? bn->g : nullptr;
  const float* b2 = bn ? bn->b : nullptr;
  const float* mu = bn ? bn->m : nullptr;
  const float* vv = bn ? bn->v : nullptr;
  if ((M & 15) == 0 && (N % 64) == 0) {
    long tiles = (M >> 4) * (N / 64);
    gemm_wmma_aligned<4><<<(int)cdiv(tiles, 8), 256, 0, s>>>(
        X, W, bias, g, b2, mu, vv, relu, Y, (int)M, N, K);
  } else if ((M & 15) == 0 && (N % 32) == 0) {
    long tiles = (M >> 4) * (N / 32);
    gemm_wmma_aligned<2><<<(int)cdiv(tiles, 8), 256, 0, s>>>(
        X, W, bias, g, b2, mu, vv, relu, Y, (int)M, N, K);
  } else {
    long tiles = ((M + 15) >> 4) * ((N + 15) >> 4);
    gemm_wmma_generic<<<(int)cdiv(tiles, 8), 256, 0, s>>>(
        X, W, bias, g, b2, mu, vv, relu, Y, (int)M, N, K);
  }
}

struct Bufs { float *FB, *FQ, *FK, *FV, *FT, *BA, *BB, *BC, *RELB; int *idxL, *idxTD, *fpsIdx; };

static void run_block(hipStream_t s, const Block& P, const float* pos, float* h,
                      int B, int N, int C, const unsigned char* mask, Bufs& u)
{
  long M2 = (long)B * N;
  long M4 = M2 * KNBR;
  // y = relu(bn1(l1 x))
  gemm(s, h, P.l1.w, nullptr, &P.bn1, 1, u.FB, M2, C, C);
  // q,k,v
  gemm(s, u.FB, P.tr.q.w, P.tr.q.b, nullptr, 0, u.FQ, M2, C, C);
  gemm(s, u.FB, P.tr.k.w, P.tr.k.b, nullptr, 0, u.FK, M2, C, C);
  gemm(s, u.FB, P.tr.v.w, P.tr.v.b, nullptr, 0, u.FV, M2, C, C);
  // layer KNN (no mask) + relative positions
  knn_kernel<<<(int)cdiv(M2, 256), 256, 0, s>>>(pos, pos, nullptr, u.idxL, B, N, N);
  rel_kernel<<<(int)cdiv(M4, 256), 256, 0, s>>>(pos, u.idxL, u.RELB, N, M4);
  // r = pe2(relu(bn(pe1(rel))))
  gemm(s, u.RELB, P.tr.pe1.w, nullptr, &P.tr.pe_bn1, 1, u.BA, M4, C, 3);
  gemm(s, u.BA, P.tr.pe2.w, P.tr.pe2.b, nullptr, 0, u.BB, M4, C, C);     // r -> BB
  // a = relu(bn(q - n_k + r)); at1; relu(bn); at2
  abuild_kernel<<<(int)cdiv(M4 * C, 256), 256, 0, s>>>(
      u.FQ, u.FK, u.BB, u.idxL,
      P.tr.at_bn1.g, P.tr.at_bn1.b, P.tr.at_bn1.m, P.tr.at_bn1.v,
      u.BA, N, C, M4 * C);
  gemm(s, u.BA, P.tr.at1.w, nullptr, &P.tr.at_bn2, 1, u.BC, M4, C, C);
  gemm(s, u.BC, P.tr.at2.w, P.tr.at2.b, nullptr, 0, u.BA, M4, C, C);
  // softmax over K + weighted sum of (n_v + r)
  softmax_agg_kernel<<<(int)cdiv(M2 * C, 256), 256, 0, s>>>(
      u.BA, u.FV, u.BB, u.idxL, mask, u.FT, N, C, M2 * C);
  // y = relu(bn(l2 y_tr)); h = relu(y + x)
  gemm(s, u.FT, P.l2.w, nullptr, &P.bn, 1, u.FB, M2, C, C);
  residual_relu_kernel<<<(int)cdiv(M2 * C, 256), 256, 0, s>>>(u.FB, h, h, M2 * C);
}

static void run_td(hipStream_t s, const Td& T, const float* p1, const float* h,
                   int B, int N, int M, int Cin, int Cout,
                   const unsigned char* mask, float* p2, float* newh, Bufs& u)
{
  fps_kernel<<<B, 256, 0, s>>>(p1, u.fpsIdx, N, M);
  gather_pts_kernel<<<(int)cdiv((long)B * M * 3, 256), 256, 0, s>>>(
      p1, u.fpsIdx, p2, N, M, (long)B * M * 3);
  knn_kernel<<<(int)cdiv((long)B * M, 256), 256, 0, s>>>(p1, p2, mask, u.idxTD, B, N, M);
  long rows = (long)B * M * KNBR;
  tdfeat_kernel<<<(int)cdiv(rows * (3 + Cin), 256), 256, 0, s>>>(
      p1, p2, h, u.idxTD, u.BA, N, M, Cin, rows * (3 + Cin));
  gemm(s, u.BA, T.l1.w, nullptr, &T.bn1, 1, u.BB, rows, Cout, Cin + 3);
  gemm(s, u.BB, T.l2.w, nullptr, &T.bn2, 1, u.BC, rows, Cout, Cout);
  maxpool_kernel<<<(int)cdiv((long)B * M * Cout, 256), 256, 0, s>>>(
      u.BC, newh, M, Cout, (long)B * M * Cout);
}

extern "C" void kernel_launch(void* const* d_in, const int* in_sizes, int n_in,
                              void* d_out, int out_size, void* d_ws, size_t ws_size,
                              hipStream_t stream)
{
  (void)in_sizes; (void)n_in; (void)out_size; (void)ws_size;
  const float* x = (const float*)d_in[0];   // [8, 2048, 3]

  Cursor cur{d_in, 1};
  Lin in1 = getLin(cur, false); Bn in_bn1 = getBn(cur);
  Lin in2 = getLin(cur, false); Bn in_bn2 = getBn(cur);
  Block blk[5]; Td td[5];
  blk[0] = getBlock(cur);
  for (int i = 1; i < 5; ++i) { td[i] = getTd(cur); blk[i] = getBlock(cur); }
  Lin fc1 = getLin(cur, true);
  Lin fc2 = getLin(cur, true);

  const int B = 8;
  const int Ns[5] = {2048, 512, 128, 32, 8};
  const int Cs[5] = {32, 64, 128, 256, 512};

  // ---- workspace carve (bump allocator, 256B aligned) ----
  char* w = (char*)d_ws;
  auto alloc = [&](size_t bytes) -> void* {
    void* p = (void*)w; w += (bytes + 255) & ~(size_t)255; return p;
  };
  float* P[5]; P[0] = nullptr;  // positions (level 0 = x itself)
  for (int s2 = 1; s2 < 5; ++s2) P[s2] = (float*)alloc((size_t)B * Ns[s2] * 3 * 4);
  unsigned char* mask0 = (unsigned char*)alloc((size_t)B * 2048);
  Bufs u;
  u.idxL   = (int*)alloc((size_t)B * 2048 * KNBR * 4);
  u.idxTD  = (int*)alloc((size_t)B * 512 * KNBR * 4);
  u.fpsIdx = (int*)alloc((size_t)B * 512 * 4);
  const size_t FEAT = (size_t)B * 2048 * 32;               // 524288 f
  float* HA = (float*)alloc(FEAT * 4);
  float* HB = (float*)alloc(FEAT * 4);
  u.FB = (float*)alloc(FEAT * 4);
  u.FQ = (float*)alloc(FEAT * 4);
  u.FK = (float*)alloc(FEAT * 4);
  u.FV = (float*)alloc(FEAT * 4);
  u.FT = (float*)alloc(FEAT * 4);
  const size_t BIG = (size_t)B * 2048 * KNBR * 32;         // 8.4M f (stage-0 worst case)
  u.BA = (float*)alloc(BIG * 4);
  u.BB = (float*)alloc(BIG * 4);
  u.BC = (float*)alloc(BIG * 4);
  u.RELB = (float*)alloc((size_t)B * 2048 * KNBR * 3 * 4);
  float* POOL  = (float*)alloc((size_t)B * 512 * 4);
  float* FC1O  = (float*)alloc((size_t)B * 256 * 4);

  // ---- forward ----
  mask_kernel<<<(int)cdiv((long)B * 2048, 256), 256, 0, stream>>>(x, mask0, (long)B * 2048);

  // input MLP: relu(bn(in2(relu(bn(in1(x))))))
  gemm(stream, x,    in1.w, nullptr, &in_bn1, 1, u.FB, (long)B * 2048, 32, 3);
  gemm(stream, u.FB, in2.w, nullptr, &in_bn2, 1, HA,   (long)B * 2048, 32, 32);

  float* h = HA;
  float* hspare = HB;
  const float* pos = x;
  const unsigned char* mask = mask0;

  run_block(stream, blk[0], pos, h, B, Ns[0], Cs[0], mask, u);

  for (int s2 = 1; s2 < 5; ++s2) {
    run_td(stream, td[s2], pos, h, B, Ns[s2 - 1], Ns[s2], Cs[s2 - 1], Cs[s2],
           mask, P[s2], hspare, u);
    { float* t = h; h = hspare; hspare = t; }
    pos = P[s2];
    mask = nullptr;                         // transition_down emits all-true mask
    run_block(stream, blk[s2], pos, h, B, Ns[s2], Cs[s2], nullptr, u);
  }

  // masked mean pool (mask all-true at this point) + classifier head
  avgpool_kernel<<<(int)cdiv((long)B * 512, 256), 256, 0, stream>>>(
      h, nullptr, POOL, Ns[4], Cs[4], (long)B * 512);
  gemm(stream, POOL, fc1.w, fc1.b, nullptr, 1, FC1O, B, 256, 512);
  gemm(stream, FC1O, fc2.w, fc2.b, nullptr, 0, (float*)d_out, B, 10, 256);
}